// MedianFilterTransform_13426067767585
// MI455X (gfx1250) — compile-verified
//
#include <hip/hip_runtime.h>
#include <hip/hip_bf16.h>

// 3x3 median blur, zero-padded, fp32, 24 planes of 1024x1024.
// Memory-bound (192 MB @ 23.3 TB/s ~ 8.3 us floor). Data path: async
// global->LDS staging (ASYNCcnt) + v_med3/v_min3/v_max3 median network
// (7 ops/pixel via rolling row-sorted triples) + non-temporal output stores.
// Tile 256x32 -> halo read overhead 1.07x. Staging specialized for
// y-interior blocks (94%): branchless, all-async, immediate offsets.

#define IMG_H 1024
#define IMG_W 1024
#define TILE_W 256
#define TILE_H 32
#define LROWS (TILE_H + 2)   // 34 rows incl. halo
#define LCOLS (TILE_W + 2)   // 258 cols incl. halo
#define LSTRIDE 264          // padded LDS row stride (floats)

typedef __attribute__((address_space(3))) int lds_int_t;   // "__shared__ int"
typedef __attribute__((address_space(1))) int glob_int_t;  // "__device__ int"

#if __has_builtin(__builtin_amdgcn_global_load_async_to_lds_b32)
#define USE_ASYNC_LDS 1
#else
#define USE_ASYNC_LDS 0
#endif

#if __has_builtin(__builtin_amdgcn_s_wait_asynccnt)
#define WAIT_ASYNC0() __builtin_amdgcn_s_wait_asynccnt(0)
#else
#define WAIT_ASYNC0() asm volatile("s_wait_asynccnt 0" ::: "memory")
#endif

__device__ __forceinline__ void async_cp(const float* gp, float* lp) {
#if USE_ASYNC_LDS
    // global_load_async_to_lds_b32 (per-lane, EXEC-masked), ASYNCcnt-tracked
    __builtin_amdgcn_global_load_async_to_lds_b32(
        (glob_int_t*)(uintptr_t)gp, (lds_int_t*)lp, /*imm off*/ 0, /*cpol*/ 0);
#else
    *lp = *gp;
#endif
}

__device__ __forceinline__ void stage_one(const float* gp, float* lp, bool inb) {
    if (inb) async_cp(gp, lp);
    else     *lp = 0.0f;   // zero padding: disjoint addresses from async writes
}

__device__ __forceinline__ float min3f(float a, float b, float c) {
    return fminf(fminf(a, b), c);      // v_min3_f32
}
__device__ __forceinline__ float max3f(float a, float b, float c) {
    return fmaxf(fmaxf(a, b), c);      // v_max3_f32
}
__device__ __forceinline__ float med3f(float a, float b, float c) {
    return __builtin_amdgcn_fmed3f(a, b, c);   // v_med3_f32
}

__global__ __launch_bounds__(256)
void median3x3_kernel(const float* __restrict__ in, float* __restrict__ out)
{
    __shared__ float tile[LROWS * LSTRIDE];

    const int tid   = threadIdx.x;
    const int tx0   = blockIdx.x * TILE_W;
    const int ty0   = blockIdx.y * TILE_H;
    const int plane = blockIdx.z;

    const float* pin  = in  + (size_t)plane * IMG_H * IMG_W;
    float*       pout = out + (size_t)plane * IMG_H * IMG_W;

    // ---- stage (TILE_H+2) x (TILE_W+2) zero-padded window into LDS ----
    // Interior cols: LDS col tid+1 == global col tx0+tid (aligned, coalesced).
    const bool y_interior = (ty0 >= 1) && (ty0 + TILE_H < IMG_H);

    if (y_interior) {
        // Hot path (94% of blocks): no bounds checks, pure async issue.
        // Row offsets are compile-time constants -> folded into immediates.
        const float* gbase = pin + (size_t)(ty0 - 1) * IMG_W + (tx0 + tid);
        float*       lbase = &tile[tid + 1];
        #pragma unroll
        for (int r = 0; r < LROWS; ++r)
            async_cp(gbase + (size_t)r * IMG_W, lbase + r * LSTRIDE);
    } else {
        #pragma unroll
        for (int r = 0; r < LROWS; ++r) {
            const int gy = ty0 - 1 + r;
            stage_one(pin + (size_t)gy * IMG_W + (tx0 + tid),
                      &tile[r * LSTRIDE + tid + 1],
                      (unsigned)gy < IMG_H);
        }
    }
    // Halo columns 0 and LCOLS-1: 2*LROWS = 68 elements, first 68 threads.
    if (tid < 2 * LROWS) {
        const int r  = tid >> 1;
        const int c  = (tid & 1) ? (LCOLS - 1) : 0;
        const int gy = ty0 - 1 + r;
        const int gx = tx0 - 1 + c;                    // tx0-1 or tx0+256
        stage_one(pin + (size_t)gy * IMG_W + gx,
                  &tile[r * LSTRIDE + c],
                  ((unsigned)gy < IMG_H) && ((unsigned)gx < IMG_W));
    }
#if USE_ASYNC_LDS
    WAIT_ASYNC0();        // each wave drains its own async loads pre-barrier
#endif
    __syncthreads();

    // ---- median network: rolling row-sorted triples ----
    const int lc = tid + 1;

    float a = tile[0 * LSTRIDE + lc - 1];
    float b = tile[0 * LSTRIDE + lc    ];
    float c = tile[0 * LSTRIDE + lc + 1];
    float lo0 = min3f(a, b, c), mid0 = med3f(a, b, c), hi0 = max3f(a, b, c);

    a = tile[1 * LSTRIDE + lc - 1];
    b = tile[1 * LSTRIDE + lc    ];
    c = tile[1 * LSTRIDE + lc + 1];
    float lo1 = min3f(a, b, c), mid1 = med3f(a, b, c), hi1 = max3f(a, b, c);

    #pragma unroll
    for (int r = 0; r < TILE_H; ++r) {
        a = tile[(r + 2) * LSTRIDE + lc - 1];
        b = tile[(r + 2) * LSTRIDE + lc    ];
        c = tile[(r + 2) * LSTRIDE + lc + 1];
        const float lo2 = min3f(a, b, c);
        const float mid2 = med3f(a, b, c);
        const float hi2 = max3f(a, b, c);

        const float m = med3f(max3f(lo0, lo1, lo2),
                              med3f(mid0, mid1, mid2),
                              min3f(hi0, hi1, hi2));

        // write-once output: non-temporal store, don't pollute L2
        __builtin_nontemporal_store(m, &pout[(size_t)(ty0 + r) * IMG_W + (tx0 + tid)]);

        lo0 = lo1; mid0 = mid1; hi0 = hi1;
        lo1 = lo2; mid1 = mid2; hi1 = hi2;
    }
}

extern "C" void kernel_launch(void* const* d_in, const int* in_sizes, int n_in,
                              void* d_out, int out_size, void* d_ws, size_t ws_size,
                              hipStream_t stream) {
    (void)n_in; (void)out_size; (void)d_ws; (void)ws_size;
    const float* in = (const float*)d_in[0];
    float* out = (float*)d_out;
    const int planes = in_sizes[0] / (IMG_H * IMG_W);   // 8*3 = 24
    dim3 grid(IMG_W / TILE_W, IMG_H / TILE_H, planes);
    median3x3_kernel<<<grid, dim3(256, 1, 1), 0, stream>>>(in, out);
}